// SE3Transformer_83270825935545
// MI455X (gfx1250) — compile-verified
//
#include <hip/hip_runtime.h>
#include <math.h>

#define FEAT 64
#define HEADS 8
#define HEAD_DIM 8

typedef __attribute__((ext_vector_type(2))) float v2f;
typedef __attribute__((ext_vector_type(8))) float v8f;

// ---------------------------------------------------------------------------
// Monotone float<->uint mapping so atomicMax(uint) implements float max for
// mixed-sign values.
// ---------------------------------------------------------------------------
__device__ __forceinline__ unsigned int float_to_ordered(float x) {
    unsigned int u = __float_as_uint(x);
    return (u & 0x80000000u) ? ~u : (u | 0x80000000u);
}
__device__ __forceinline__ float ordered_to_float(unsigned int m) {
    return (m & 0x80000000u) ? __uint_as_float(m & 0x7FFFFFFFu)
                             : __uint_as_float(~m);
}

// ---------------------------------------------------------------------------
// Init: zero accumulator + softmax stats (stats[0..7] = max-bits, [8..15] = sum)
// ---------------------------------------------------------------------------
__global__ void init_kernel(float* __restrict__ acc, unsigned int* __restrict__ stats,
                            int n_acc) {
    int i = blockIdx.x * blockDim.x + threadIdx.x;
    if (i < n_acc) acc[i] = 0.0f;
    if (i < 16) stats[i] = 0u;   // 0 is below ordered(-FLT_MAX); sum bits = 0.0f
}

// ---------------------------------------------------------------------------
// Y[m][n] = sum_k X[m][k] * W[n][k] + bias[n]      (i.e. Y = X @ W^T + b)
// One wave computes a 16x64 tile with V_WMMA_F32_16X16X4_F32.
//   A (16x4, MxK): lanes 0-15 -> (K0,K1), lanes 16-31 -> (K2,K3)
//   B (4x16, KxN): lanes hold column N=r; same half-split over K.
//   Since B[k][n] = W[n][k], both fragments are float2 loads at
//   base + row*64 + k0 + 2*half.
//   D (16x16): VGPR j holds row m0 + j + 8*half, column n0 + r.
// ---------------------------------------------------------------------------
#define GEMM_WAVES 4
__global__ void proj_gemm(const float* __restrict__ X, const float* __restrict__ W,
                          const float* __restrict__ bias, float* __restrict__ Y,
                          int n_rows) {
    const int wave = threadIdx.x >> 5;
    const int tile = blockIdx.x * GEMM_WAVES + wave;
    const int m0 = tile * 16;
    if (m0 + 16 > n_rows) return;               // wave-uniform exit (EXEC stays all-1)

    const int lane = threadIdx.x & 31;
    const int half = lane >> 4;                 // 0: K0/K1, 1: K2/K3
    const int r    = lane & 15;

    v8f acc0 = {}, acc1 = {}, acc2 = {}, acc3 = {};
    const float* xrow = X + (size_t)(m0 + r) * FEAT;

#pragma unroll
    for (int k0 = 0; k0 < FEAT; k0 += 4) {
        const v2f a = *(const v2f*)(xrow + k0 + 2 * half);
        const v2f b0 = *(const v2f*)(W + (size_t)(0  + r) * FEAT + k0 + 2 * half);
        const v2f b1 = *(const v2f*)(W + (size_t)(16 + r) * FEAT + k0 + 2 * half);
        const v2f b2 = *(const v2f*)(W + (size_t)(32 + r) * FEAT + k0 + 2 * half);
        const v2f b3 = *(const v2f*)(W + (size_t)(48 + r) * FEAT + k0 + 2 * half);
        acc0 = __builtin_amdgcn_wmma_f32_16x16x4_f32(false, a, false, b0, (short)0, acc0, false, false);
        acc1 = __builtin_amdgcn_wmma_f32_16x16x4_f32(false, a, false, b1, (short)0, acc1, false, false);
        acc2 = __builtin_amdgcn_wmma_f32_16x16x4_f32(false, a, false, b2, (short)0, acc2, false, false);
        acc3 = __builtin_amdgcn_wmma_f32_16x16x4_f32(false, a, false, b3, (short)0, acc3, false, false);
    }

    const float bb0 = bias[0 + r], bb1 = bias[16 + r], bb2 = bias[32 + r], bb3 = bias[48 + r];
#pragma unroll
    for (int j = 0; j < 8; ++j) {
        float* yrow = Y + (size_t)(m0 + j + 8 * half) * FEAT;
        yrow[0  + r] = acc0[j] + bb0;
        yrow[16 + r] = acc1[j] + bb1;
        yrow[32 + r] = acc2[j] + bb2;
        yrow[48 + r] = acc3[j] + bb3;
    }
}

// ---------------------------------------------------------------------------
// Per-(edge,head) attention logit:
//   logit = (q[row,h,:] . k[col,h,:]) / sqrt(8) * sigmoid(dist * Wa[h] + ba[h])
// Also reduces the per-head global max (LDS pre-reduction -> global atomicMax).
// ---------------------------------------------------------------------------
__global__ void edge_logits(const int* __restrict__ rows, const int* __restrict__ cols,
                            const float* __restrict__ Q, const float* __restrict__ Kf,
                            const float* __restrict__ coords,
                            const float* __restrict__ Wa, const float* __restrict__ ba,
                            float* __restrict__ logits,
                            unsigned int* __restrict__ head_max_bits, int n_pairs) {
    __shared__ unsigned int smax[HEADS];
    if (threadIdx.x < HEADS) smax[threadIdx.x] = 0u;
    __syncthreads();

    const int i = blockIdx.x * blockDim.x + threadIdx.x;
    if (i < n_pairs) {
        const int e = i >> 3;
        const int h = i & 7;
        const int row = rows[e];
        const int col = cols[e];

        const float4* qp = (const float4*)(Q  + (size_t)row * FEAT + h * HEAD_DIM);
        const float4* kp = (const float4*)(Kf + (size_t)col * FEAT + h * HEAD_DIM);
        const float4 q0 = qp[0], q1 = qp[1];
        const float4 k0 = kp[0], k1 = kp[1];
        float dot = q0.x * k0.x + q0.y * k0.y + q0.z * k0.z + q0.w * k0.w +
                    q1.x * k1.x + q1.y * k1.y + q1.z * k1.z + q1.w * k1.w;
        dot *= 0.35355339059327373f;            // 1/sqrt(8)

        const float dx = coords[row * 3 + 0] - coords[col * 3 + 0];
        const float dy = coords[row * 3 + 1] - coords[col * 3 + 1];
        const float dz = coords[row * 3 + 2] - coords[col * 3 + 2];
        const float dist = sqrtf(dx * dx + dy * dy + dz * dz);
        const float z = dist * Wa[h] + ba[h];
        const float gate = 1.0f / (1.0f + __expf(-z));

        const float logit = dot * gate;
        logits[i] = logit;
        atomicMax(&smax[h], float_to_ordered(logit));
    }
    __syncthreads();
    if (threadIdx.x < HEADS) atomicMax(&head_max_bits[threadIdx.x], smax[threadIdx.x]);
}

// ---------------------------------------------------------------------------
// p = exp(logit - head_max); accumulate per-head sum (LDS ds_add_f32 -> global).
// ---------------------------------------------------------------------------
__global__ void softmax_norm(float* __restrict__ logits,
                             const unsigned int* __restrict__ head_max_bits,
                             float* __restrict__ head_sum, int n_pairs) {
    __shared__ float ssum[HEADS];
    if (threadIdx.x < HEADS) ssum[threadIdx.x] = 0.0f;
    __syncthreads();

    const int i = blockIdx.x * blockDim.x + threadIdx.x;
    if (i < n_pairs) {
        const int h = i & 7;
        const float mx = ordered_to_float(head_max_bits[h]);
        const float p = __expf(logits[i] - mx);
        logits[i] = p;
        atomicAdd(&ssum[h], p);
    }
    __syncthreads();
    if (threadIdx.x < HEADS) atomicAdd(&head_sum[threadIdx.x], ssum[threadIdx.x]);
}

// ---------------------------------------------------------------------------
// Scatter: acc[row, h, :] += (p / head_sum[h]) * V[col, h, :]
// ---------------------------------------------------------------------------
__global__ void edge_scatter(const int* __restrict__ rows, const int* __restrict__ cols,
                             const float* __restrict__ logits,
                             const float* __restrict__ head_sum,
                             const float* __restrict__ Vf, float* __restrict__ acc,
                             int n_pairs) {
    const int i = blockIdx.x * blockDim.x + threadIdx.x;
    if (i >= n_pairs) return;
    const int e = i >> 3;
    const int h = i & 7;
    const int row = rows[e];
    const int col = cols[e];
    const float w = logits[i] / head_sum[h];

    const float4* vp = (const float4*)(Vf + (size_t)col * FEAT + h * HEAD_DIM);
    const float4 v0 = vp[0], v1 = vp[1];
    float* op = acc + (size_t)row * FEAT + h * HEAD_DIM;
    atomicAdd(op + 0, w * v0.x);
    atomicAdd(op + 1, w * v0.y);
    atomicAdd(op + 2, w * v0.z);
    atomicAdd(op + 3, w * v0.w);
    atomicAdd(op + 4, w * v1.x);
    atomicAdd(op + 5, w * v1.y);
    atomicAdd(op + 6, w * v1.z);
    atomicAdd(op + 7, w * v1.w);
}

// ---------------------------------------------------------------------------
extern "C" void kernel_launch(void* const* d_in, const int* in_sizes, int n_in,
                              void* d_out, int out_size, void* d_ws, size_t ws_size,
                              hipStream_t stream) {
    const float* features = (const float*)d_in[0];
    const float* coords   = (const float*)d_in[1];
    const float* Wq = (const float*)d_in[2];
    const float* bq = (const float*)d_in[3];
    const float* Wk = (const float*)d_in[4];
    const float* bk = (const float*)d_in[5];
    const float* Wv = (const float*)d_in[6];
    const float* bv = (const float*)d_in[7];
    const float* Wa = (const float*)d_in[8];
    const float* ba = (const float*)d_in[9];
    const float* Wo = (const float*)d_in[10];
    const float* bo = (const float*)d_in[11];
    const int*   edge = (const int*)d_in[12];

    const int N = in_sizes[0] / FEAT;          // 50000
    const int E = in_sizes[12] / 2;            // 800000
    const int* rows = edge;
    const int* cols = edge + E;

    // Workspace layout
    char* w = (char*)d_ws;
    const size_t nodeBytes = (size_t)N * FEAT * sizeof(float);
    float* Q   = (float*)(w);
    float* K   = (float*)(w + nodeBytes);
    float* V   = (float*)(w + 2 * nodeBytes);
    float* ACC = (float*)(w + 3 * nodeBytes);
    float* LOG = (float*)(w + 4 * nodeBytes);
    unsigned int* STATS = (unsigned int*)(w + 4 * nodeBytes + (size_t)E * HEADS * sizeof(float));
    unsigned int* head_max_bits = STATS;
    float*        head_sum      = (float*)(STATS + HEADS);

    // 0) init accumulator + stats
    {
        const int n_acc = N * FEAT;
        init_kernel<<<(n_acc + 255) / 256, 256, 0, stream>>>(ACC, STATS, n_acc);
    }

    // 1) Q/K/V projections (WMMA f32 16x16x4)
    {
        const int tiles = (N + 15) / 16;
        const int gx = (tiles + GEMM_WAVES - 1) / GEMM_WAVES;
        proj_gemm<<<gx, 32 * GEMM_WAVES, 0, stream>>>(features, Wq, bq, Q, N);
        proj_gemm<<<gx, 32 * GEMM_WAVES, 0, stream>>>(features, Wk, bk, K, N);
        proj_gemm<<<gx, 32 * GEMM_WAVES, 0, stream>>>(features, Wv, bv, V, N);
    }

    const int n_pairs = E * HEADS;
    const int pgrid = (n_pairs + 255) / 256;

    // 2) logits + global per-head max
    edge_logits<<<pgrid, 256, 0, stream>>>(rows, cols, Q, K, coords, Wa, ba,
                                           LOG, head_max_bits, n_pairs);
    // 3) exp + global per-head sum
    softmax_norm<<<pgrid, 256, 0, stream>>>(LOG, head_max_bits, head_sum, n_pairs);
    // 4) scatter-add weighted values
    edge_scatter<<<pgrid, 256, 0, stream>>>(rows, cols, LOG, head_sum, V, ACC, n_pairs);

    // 5) output projection (WMMA)
    {
        const int tiles = (N + 15) / 16;
        const int gx = (tiles + GEMM_WAVES - 1) / GEMM_WAVES;
        proj_gemm<<<gx, 32 * GEMM_WAVES, 0, stream>>>(ACC, Wo, bo, (float*)d_out, N);
    }
}